// _PointNet2SAModuleBase_49855980372368
// MI455X (gfx1250) — compile-verified
//
#include <hip/hip_runtime.h>

#define B_ 4
#define N_ 8192
#define NPOINT_ 2048
#define NSAMPLE_ 32
#define C_ 128
#define CIN_ 131     // 3 xyz + 128 features (reference concat order: xyz first)
#define KP_ 132      // padded K for layer1 (col 131 = 0); also LDS row pitch
#define H1_ 64
#define H2_ 128
#define OUT_ 128
#define RADIUS2_ (0.2f * 0.2f)

typedef float v2f __attribute__((ext_vector_type(2)));
typedef float v8f __attribute__((ext_vector_type(8)));

// ---------------------------------------------------------------------------
// Kernel 0: transpose features (B,C,N) -> featT (B,N,C) so gathers are
// contiguous 512B rows instead of 128 stride-32KB scalars.
// ---------------------------------------------------------------------------
__global__ __launch_bounds__(256) void pn2_transpose_kernel(
    const float* __restrict__ f, float* __restrict__ fT) {
  __shared__ float tile[32][33];
  const int b = blockIdx.z;
  const int n0 = blockIdx.x * 32, c0 = blockIdx.y * 32;
  const int tx = threadIdx.x & 31;
  const int ty = threadIdx.x >> 5;
  for (int j = ty; j < 32; j += 8)
    tile[j][tx] = f[((size_t)b * C_ + (c0 + j)) * N_ + n0 + tx];
  __syncthreads();
  for (int j = ty; j < 32; j += 8)
    fT[((size_t)b * N_ + n0 + j) * C_ + c0 + tx] = tile[tx][j];
}

// ---------------------------------------------------------------------------
// Kernel 0b: pre-pack a weight matrix into WMMA B-fragment order so the GEMM
// hot loops do single coalesced 256B b64 loads with no guards/divergence.
// Fragment (kk, nt): lane holds B[k..k+1][n] with n = nt*16 + (lane&15),
// k = kk*4 + (lane>>4)*2.  For layer1 (remap=1) the A matrix is stored as
// [feat(128)|xyz(3)|0], so map packed col c -> W1 col (c<128 ? c+3 : c-128),
// and zero-fill c >= 131.
// ---------------------------------------------------------------------------
__global__ __launch_bounds__(256) void pn2_pack_kernel(
    const float* __restrict__ W, float* __restrict__ dst, int ncols, int nkk,
    int nnt, int remap) {
  const int tid = blockIdx.x * 256 + threadIdx.x;
  const int total = nkk * nnt * 32;
  if (tid >= total) return;
  const int lane = tid & 31;
  const int nt = (tid >> 5) % nnt;
  const int kk = tid / (32 * nnt);
  const int ln = lane & 15;
  const int khalf = (lane >> 4) * 2;
  const int ks = kk * 4 + khalf;
  const int n = nt * 16 + ln;
  float v0, v1;
  if (remap) {
    const int c0 = ks, c1 = ks + 1;
    v0 = (c0 < CIN_) ? W[n * CIN_ + ((c0 < C_) ? (c0 + 3) : (c0 - C_))] : 0.0f;
    v1 = (c1 < CIN_) ? W[n * CIN_ + ((c1 < C_) ? (c1 + 3) : (c1 - C_))] : 0.0f;
  } else {
    v0 = W[n * ncols + ks];
    v1 = W[n * ncols + ks + 1];
  }
  dst[(size_t)tid * 2 + 0] = v0;
  dst[(size_t)tid * 2 + 1] = v1;
}

// ---------------------------------------------------------------------------
// Kernel 1: farthest point sampling. One block (1024 thr = 32 waves) per
// batch; each thread keeps its 8 points + running min-dist in REGISTERS so
// the 2047-iteration serial loop touches global memory only for the 3-float
// broadcast of the newly chosen center. Tie-break to lowest index matches
// jnp.argmax first-occurrence semantics.
// ---------------------------------------------------------------------------
__global__ __launch_bounds__(1024) void pn2_fps_kernel(
    const float* __restrict__ xyz, int* __restrict__ center_idx,
    float* __restrict__ new_xyz) {
  const int b = blockIdx.x;
  const int t = threadIdx.x;
  const int lane = t & 31, wid = t >> 5;
  const float* xb = xyz + (size_t)b * N_ * 3;

  float px[8], py[8], pz[8], dist[8];
#pragma unroll
  for (int i = 0; i < 8; ++i) {
    const int p = t + i * 1024;
    px[i] = xb[p * 3 + 0];
    py[i] = xb[p * 3 + 1];
    pz[i] = xb[p * 3 + 2];
    dist[i] = 1e10f;
  }

  __shared__ float swv[32];
  __shared__ int swi[32];
  __shared__ float s_cx, s_cy, s_cz;

  if (t == 0) {
    center_idx[b * NPOINT_ + 0] = 0;
    s_cx = xb[0];
    s_cy = xb[1];
    s_cz = xb[2];
    new_xyz[((size_t)b * NPOINT_ + 0) * 3 + 0] = xb[0];
    new_xyz[((size_t)b * NPOINT_ + 0) * 3 + 1] = xb[1];
    new_xyz[((size_t)b * NPOINT_ + 0) * 3 + 2] = xb[2];
  }
  __syncthreads();

  for (int it = 1; it < NPOINT_; ++it) {
    const float cx = s_cx, cy = s_cy, cz = s_cz;
    float bv = -1.0f;
    int bi = 0;
#pragma unroll
    for (int i = 0; i < 8; ++i) {
      const float dx = px[i] - cx, dy = py[i] - cy, dz = pz[i] - cz;
      float d = dx * dx + dy * dy + dz * dz;
      d = fminf(dist[i], d);
      dist[i] = d;
      if (d > bv) {  // ascending local indices -> '>' keeps lowest on tie
        bv = d;
        bi = t + i * 1024;
      }
    }
    // wave32 reduction
#pragma unroll
    for (int off = 16; off > 0; off >>= 1) {
      const float ov = __shfl_down(bv, off, 32);
      const int oi = __shfl_down(bi, off, 32);
      if (ov > bv || (ov == bv && oi < bi)) {
        bv = ov;
        bi = oi;
      }
    }
    if (lane == 0) {
      swv[wid] = bv;
      swi[wid] = bi;
    }
    __syncthreads();
    if (wid == 0) {
      bv = swv[lane];
      bi = swi[lane];
#pragma unroll
      for (int off = 16; off > 0; off >>= 1) {
        const float ov = __shfl_down(bv, off, 32);
        const int oi = __shfl_down(bi, off, 32);
        if (ov > bv || (ov == bv && oi < bi)) {
          bv = ov;
          bi = oi;
        }
      }
      if (lane == 0) {
        center_idx[b * NPOINT_ + it] = bi;
        const float nx = xb[bi * 3 + 0], ny = xb[bi * 3 + 1],
                    nz = xb[bi * 3 + 2];
        s_cx = nx;
        s_cy = ny;
        s_cz = nz;
        new_xyz[((size_t)b * NPOINT_ + it) * 3 + 0] = nx;
        new_xyz[((size_t)b * NPOINT_ + it) * 3 + 1] = ny;
        new_xyz[((size_t)b * NPOINT_ + it) * 3 + 2] = nz;
      }
    }
    __syncthreads();
  }
}

// ---------------------------------------------------------------------------
// Kernel 2: ball query. One wave32 per center; scan 8192 points in chunks of
// 32, __ballot + prefix popcount appends in-order the first 32 indices within
// radius (early exit). Pad with first hit (center itself is always a hit).
// ---------------------------------------------------------------------------
__global__ __launch_bounds__(256) void pn2_ballq_kernel(
    const float* __restrict__ xyz, const float* __restrict__ new_xyz,
    int* __restrict__ g_idx) {
  const int lane = threadIdx.x & 31, wid = threadIdx.x >> 5;
  const int center = blockIdx.x * 8 + wid;
  const int b = center / NPOINT_;
  const float* xb = xyz + (size_t)b * N_ * 3;
  const float cx = new_xyz[(size_t)center * 3 + 0];
  const float cy = new_xyz[(size_t)center * 3 + 1];
  const float cz = new_xyz[(size_t)center * 3 + 2];

  __shared__ int buf[8][NSAMPLE_];
  int cnt = 0;
  for (int base = 0; base < N_ && cnt < NSAMPLE_; base += 32) {
    const int p = base + lane;
    const float dx = xb[p * 3 + 0] - cx;
    const float dy = xb[p * 3 + 1] - cy;
    const float dz = xb[p * 3 + 2] - cz;
    const float d2 = dx * dx + dy * dy + dz * dz;
    const bool in = d2 <= RADIUS2_;
    const unsigned mask = (unsigned)__ballot(in);
    const int total = __popc(mask);
    const int room = NSAMPLE_ - cnt;
    const int take = total < room ? total : room;
    const int rank = __popc(mask & ((1u << lane) - 1u));
    if (in && rank < take) buf[wid][cnt + rank] = p;
    cnt += take;
  }
  __syncthreads();
  const int first = buf[wid][0];
  g_idx[(size_t)center * NSAMPLE_ + lane] =
      (lane < cnt) ? buf[wid][lane] : first;
}

// ---------------------------------------------------------------------------
// Kernel 3: fused gather + MLP(131->64 relu, 64->128 relu) + max-pool over
// the 32 samples, using v_wmma_f32_16x16x4_f32 (full f32 precision).
// Block = 64 thr = 2 waves, one center per wave. A (32x132) staged in LDS
// with layout [feat(128) | xyz_rel(3) | 0]; layer-1 output h1 (relu) is
// written back over the just-consumed half of A (union trick: 34KB LDS).
// B fragments come pre-packed (coalesced b64, no guards).
// ---------------------------------------------------------------------------
__global__ __launch_bounds__(64) void pn2_group_mlp_kernel(
    const float* __restrict__ xyz, const float* __restrict__ featT,
    const float* __restrict__ new_xyz, const int* __restrict__ g_idx,
    const float* __restrict__ Bp1, const float* __restrict__ b1,
    const float* __restrict__ Bp2, const float* __restrict__ b2,
    float* __restrict__ pooledOut) {
  const int lane = threadIdx.x & 31, wid = threadIdx.x >> 5;
  const int center = blockIdx.x * 2 + wid;
  const int b = center / NPOINT_;

  __shared__ float Ain[2][NSAMPLE_ * KP_];
  float* A = &Ain[wid][0];

  const float cx = new_xyz[(size_t)center * 3 + 0];
  const float cy = new_xyz[(size_t)center * 3 + 1];
  const float cz = new_xyz[(size_t)center * 3 + 2];

  // ---- gather: 32 rows, each row one coalesced 512B feature read ----
  for (int r = 0; r < NSAMPLE_; ++r) {
    const int idx = g_idx[(size_t)center * NSAMPLE_ + r];
    const float4 f4 =
        *(const float4*)(featT + ((size_t)b * N_ + idx) * C_ + lane * 4);
    *(float4*)(A + r * KP_ + lane * 4) = f4;
    if (lane < 3) {
      const float cv = (lane == 0) ? cx : ((lane == 1) ? cy : cz);
      A[r * KP_ + 128 + lane] = xyz[((size_t)b * N_ + idx) * 3 + lane] - cv;
    }
    if (lane == 3) A[r * KP_ + 131] = 0.0f;
  }
  __syncthreads();

  const int ln = lane & 15;

  // ---- layer 1: 32x132 @ 132x64 ----
  float b1v[4];
#pragma unroll
  for (int nt = 0; nt < 4; ++nt) b1v[nt] = b1[nt * 16 + ln];

  for (int mt = 0; mt < 2; ++mt) {
    v8f acc[4] = {};
    const int arow = mt * 16 + ln;
    for (int kk = 0; kk < 33; ++kk) {
      const int ks = kk * 4 + ((lane >> 4) * 2);
      v2f a;
      a.x = A[arow * KP_ + ks];
      a.y = A[arow * KP_ + ks + 1];
#pragma unroll
      for (int nt = 0; nt < 4; ++nt) {
        const v2f bb =
            *(const v2f*)(Bp1 + ((size_t)(kk * 4 + nt) * 32 + lane) * 2);
        acc[nt] = __builtin_amdgcn_wmma_f32_16x16x4_f32(
            false, a, false, bb, (short)0, acc[nt], false, false);
      }
    }
    // epilogue: bias+relu, store h1 over the consumed rows of A
#pragma unroll
    for (int nt = 0; nt < 4; ++nt) {
#pragma unroll
      for (int i = 0; i < 8; ++i) {
        float v = acc[nt][i] + b1v[nt];
        v = fmaxf(v, 0.0f);
        const int row = mt * 16 + i + ((lane >> 4) * 8);
        A[row * KP_ + nt * 16 + ln] = v;
      }
    }
  }
  __syncthreads();

  // ---- layer 2: 32x64 @ 64x128, fused bias+relu+maxpool ----
  float b2v[8], pooled[8];
#pragma unroll
  for (int nt = 0; nt < 8; ++nt) {
    b2v[nt] = b2[nt * 16 + ln];
    pooled[nt] = 0.0f;  // relu output >= 0, so 0 is the identity for the max
  }
  for (int mt = 0; mt < 2; ++mt) {
    v8f acc[8] = {};
    const int arow = mt * 16 + ln;
    for (int kk = 0; kk < 16; ++kk) {
      const int ks = kk * 4 + ((lane >> 4) * 2);
      v2f a;
      a.x = A[arow * KP_ + ks];
      a.y = A[arow * KP_ + ks + 1];
#pragma unroll
      for (int nt = 0; nt < 8; ++nt) {
        const v2f bb =
            *(const v2f*)(Bp2 + ((size_t)(kk * 8 + nt) * 32 + lane) * 2);
        acc[nt] = __builtin_amdgcn_wmma_f32_16x16x4_f32(
            false, a, false, bb, (short)0, acc[nt], false, false);
      }
    }
#pragma unroll
    for (int nt = 0; nt < 8; ++nt) {
      float m = 0.0f;
#pragma unroll
      for (int i = 0; i < 8; ++i) m = fmaxf(m, acc[nt][i] + b2v[nt]);
      pooled[nt] = fmaxf(pooled[nt], m);
    }
  }
  // columns are split rows 0-7 (lane n) / rows 8-15 (lane n+16): merge halves
#pragma unroll
  for (int nt = 0; nt < 8; ++nt) {
    const float o = __shfl_xor(pooled[nt], 16, 32);
    pooled[nt] = fmaxf(pooled[nt], o);
  }
  if (lane < 16) {
#pragma unroll
    for (int nt = 0; nt < 8; ++nt)
      pooledOut[(size_t)center * H2_ + nt * 16 + lane] = pooled[nt];
  }
}

// ---------------------------------------------------------------------------
// Kernel 4: out = pooled(8192x128) @ Wsa^T(128x128) + bsa, written transposed
// as (B, OUT, NPOINT). WMMA f32, 4 waves x 16 rows per block. A tile staged
// in LDS (coalesced global reads, bank-padded rows); B pre-packed.
// ---------------------------------------------------------------------------
__global__ __launch_bounds__(128) void pn2_final_kernel(
    const float* __restrict__ pooled, const float* __restrict__ BpSA,
    const float* __restrict__ bsa, float* __restrict__ out) {
  __shared__ float P[64 * KP_];  // 64 rows, pitch 132 (bank-spread)
  const int lane = threadIdx.x & 31, wid = threadIdx.x >> 5;
  const int rblk = blockIdx.x * 64;

  // cooperative coalesced load of the 64x128 A tile
  for (int i = threadIdx.x; i < 64 * (H2_ / 4); i += 128) {
    const int r = i >> 5;        // row
    const int c4 = i & 31;       // float4 column
    *(float4*)(P + r * KP_ + c4 * 4) =
        *(const float4*)(pooled + ((size_t)(rblk + r)) * H2_ + c4 * 4);
  }
  __syncthreads();

  const int ln = lane & 15;
  const int arow = wid * 16 + ln;

  float bsv[8];
#pragma unroll
  for (int nt = 0; nt < 8; ++nt) bsv[nt] = bsa[nt * 16 + ln];

  v8f acc[8] = {};
  for (int kk = 0; kk < 32; ++kk) {
    const int ks = kk * 4 + ((lane >> 4) * 2);
    v2f a;
    a.x = P[arow * KP_ + ks];
    a.y = P[arow * KP_ + ks + 1];
#pragma unroll
    for (int nt = 0; nt < 8; ++nt) {
      const v2f bb =
          *(const v2f*)(BpSA + ((size_t)(kk * 8 + nt) * 32 + lane) * 2);
      acc[nt] = __builtin_amdgcn_wmma_f32_16x16x4_f32(
          false, a, false, bb, (short)0, acc[nt], false, false);
    }
  }
#pragma unroll
  for (int nt = 0; nt < 8; ++nt) {
    const int o = nt * 16 + ln;
#pragma unroll
    for (int i = 0; i < 8; ++i) {
      const int g = rblk + wid * 16 + i + ((lane >> 4) * 8);  // global row
      const int bb_ = g >> 11;                                // / NPOINT_
      const int sl = g & (NPOINT_ - 1);
      out[((size_t)bb_ * OUT_ + o) * NPOINT_ + sl] = acc[nt][i] + bsv[nt];
    }
  }
}

// ---------------------------------------------------------------------------
extern "C" void kernel_launch(void* const* d_in, const int* in_sizes, int n_in,
                              void* d_out, int out_size, void* d_ws,
                              size_t ws_size, hipStream_t stream) {
  (void)in_sizes; (void)n_in; (void)out_size; (void)ws_size;
  const float* xyz = (const float*)d_in[0];       // (B,N,3)
  const float* features = (const float*)d_in[1];  // (B,C,N)
  const float* W1 = (const float*)d_in[2];        // (64,131)
  const float* b1 = (const float*)d_in[3];
  const float* W2 = (const float*)d_in[4];        // (128,64)
  const float* b2 = (const float*)d_in[5];
  const float* Wsa = (const float*)d_in[6];       // (128,128)
  const float* bsa = (const float*)d_in[7];

  float* new_xyz = (float*)d_out;                         // (B,NPOINT,3)
  float* out = (float*)d_out + (size_t)B_ * NPOINT_ * 3;  // (B,OUT,NPOINT)

  char* ws = (char*)d_ws;
  size_t off = 0;
  int* center_idx = (int*)(ws + off);
  off += (size_t)B_ * NPOINT_ * sizeof(int);             // 32 KB
  off = (off + 255) & ~(size_t)255;
  int* g_idx = (int*)(ws + off);
  off += (size_t)B_ * NPOINT_ * NSAMPLE_ * sizeof(int);  // 1 MB
  off = (off + 255) & ~(size_t)255;
  float* pooledp = (float*)(ws + off);
  off += (size_t)B_ * NPOINT_ * H2_ * sizeof(float);     // 4 MB
  off = (off + 255) & ~(size_t)255;
  float* featT = (float*)(ws + off);
  off += (size_t)B_ * N_ * C_ * sizeof(float);           // 16 MB
  off = (off + 255) & ~(size_t)255;
  float* Bp1 = (float*)(ws + off);
  off += (size_t)33 * 4 * 32 * 2 * sizeof(float);        // 33.8 KB
  off = (off + 255) & ~(size_t)255;
  float* Bp2 = (float*)(ws + off);
  off += (size_t)16 * 8 * 32 * 2 * sizeof(float);        // 32 KB
  off = (off + 255) & ~(size_t)255;
  float* BpSA = (float*)(ws + off);
  off += (size_t)32 * 8 * 32 * 2 * sizeof(float);        // 64 KB

  // weight pre-pack (tiny, one-shot)
  pn2_pack_kernel<<<(33 * 4 * 32 + 255) / 256, 256, 0, stream>>>(
      W1, Bp1, CIN_, 33, 4, 1);
  pn2_pack_kernel<<<(16 * 8 * 32 + 255) / 256, 256, 0, stream>>>(
      W2, Bp2, H1_, 16, 8, 0);
  pn2_pack_kernel<<<(32 * 8 * 32 + 255) / 256, 256, 0, stream>>>(
      Wsa, BpSA, H2_, 32, 8, 0);

  pn2_transpose_kernel<<<dim3(N_ / 32, C_ / 32, B_), 256, 0, stream>>>(features,
                                                                       featT);
  pn2_fps_kernel<<<B_, 1024, 0, stream>>>(xyz, center_idx, new_xyz);
  pn2_ballq_kernel<<<(B_ * NPOINT_) / 8, 256, 0, stream>>>(xyz, new_xyz, g_idx);
  pn2_group_mlp_kernel<<<(B_ * NPOINT_) / 2, 64, 0, stream>>>(
      xyz, featT, new_xyz, g_idx, Bp1, b1, Bp2, b2, pooledp);
  pn2_final_kernel<<<(B_ * NPOINT_) / 64, 128, 0, stream>>>(pooledp, BpSA, bsa,
                                                            out);
}